// TtLlamaAttention_61478161875545
// MI455X (gfx1250) — compile-verified
//
#include <hip/hip_runtime.h>
#include <hip/hip_bf16.h>

// Llama-3-8B decode attention step for MI455X (gfx1250), wave32.
// All matmuls use V_WMMA_F32_16X16X4_F32 (fp32 data, memory-bound problem).
// B=4, HID=4096, S=8192, H=32, KV=8, D=128, G=4.

#define BB   4
#define HID  4096
#define SS   8192
#define HH   32
#define KVH  8
#define DD   128
#define GG   4

typedef float v2f __attribute__((ext_vector_type(2)));
typedef float v8f __attribute__((ext_vector_type(8)));

__device__ __forceinline__ v8f wmma_f32_16x16x4(v2f a, v2f b, v8f c) {
  // 8 args: (neg_a, A, neg_b, B, c_mod, C, reuse_a, reuse_b)
  return __builtin_amdgcn_wmma_f32_16x16x4_f32(false, a, false, b, (short)0, c, false, false);
}

// ---------------------------------------------------------------------------
// out[4, N] = A4[4, K] @ W[K, N]  (row-major).
// 8 waves per block: wave w accumulates K-range [w*K/8, (w+1)*K/8), then
// LDS tree-sum of the partial 16x16 C tiles. One block per 16-column tile.
// A rows 4..15 are clamped duplicates (finite); D rows 4..15 unstored.
// ---------------------------------------------------------------------------
__global__ void gemv_wmma(const float* __restrict__ A4, const float* __restrict__ W,
                          float* __restrict__ out, int K, int N) {
  const int tid  = threadIdx.x;
  const int wave = tid >> 5;                        // 0..7
  const int lane = tid & 31;
  const int half = lane >> 4;
  const int lm   = lane & 15;
  const int n0   = blockIdx.x * 16;
  const int mA   = (lm < 4) ? lm : 3;               // clamp padded rows
  const float* arow = A4 + (size_t)mA * K;

  const int kchunk = K >> 3;                        // K/8 per wave
  const int kbeg   = wave * kchunk;
  v8f c = {};
#pragma unroll 4
  for (int k = kbeg; k < kbeg + kchunk; k += 4) {
    const int ka = k + 2 * half;                    // K pair per half-wave
    v2f a = *(const v2f*)(arow + ka);               // global_load_b64
    v2f b; b.x = W[(size_t)ka * N + n0 + lm];
           b.y = W[(size_t)(ka + 1) * N + n0 + lm];
    c = wmma_f32_16x16x4(a, b, c);
  }

  __shared__ float red[8][32][8];
#pragma unroll
  for (int j = 0; j < 8; ++j) red[wave][lane][j] = c[j];
  __syncthreads();
  if (wave == 0 && half == 0) {
#pragma unroll
    for (int j = 0; j < 4; ++j) {
      float s = 0.0f;
#pragma unroll
      for (int w = 0; w < 8; ++w) s += red[w][lane][j];
      out[(size_t)j * N + n0 + lm] = s;
    }
  }
}

// ---------------------------------------------------------------------------
// RoPE: rope_out[b, r, :] = qkv[b, r, :] @ rot[b]  for rows r=0..39 (q:0..31, k:32..39).
// q (rows 0..31) and k (rows 32..39) are contiguous in the qkv buffer.
// ---------------------------------------------------------------------------
__global__ void rope_wmma(const float* __restrict__ qkv, const float* __restrict__ rot,
                          float* __restrict__ rope_out) {
  const int lane = threadIdx.x & 31;
  const int half = lane >> 4;
  const int lm   = lane & 15;
  const int n0   = blockIdx.x * 16;                 // 0..112
  const int m0   = blockIdx.y * 16;                 // 0,16,32
  const int b    = blockIdx.z;
  int r = m0 + lm; if (r > 39) r = 39;              // clamp padded rows
  const float* arow = qkv + (size_t)b * ((HH + 2 * KVH) * DD) + (size_t)r * DD;
  const float* R    = rot + (size_t)b * DD * DD;
  v8f c = {};
#pragma unroll 4
  for (int k = 0; k < DD; k += 4) {
    const int ka = k + 2 * half;
    v2f a = *(const v2f*)(arow + ka);
    v2f bv; bv.x = R[(size_t)ka * DD + n0 + lm];
            bv.y = R[(size_t)(ka + 1) * DD + n0 + lm];
    c = wmma_f32_16x16x4(a, bv, c);
  }
#pragma unroll
  for (int j = 0; j < 8; ++j) {
    const int row = m0 + j + 8 * half;
    if (row < 40)
      rope_out[(size_t)b * (40 * DD) + (size_t)row * DD + n0 + lm] = c[j];
  }
}

// ---------------------------------------------------------------------------
// scores[b,kv,g,s] = scale * (q'[b,kv,g,:] . key[s,:])  for s <= pos[b].
// key[s] = cache_k row for s<pos, new rotated k (rope rows 32..39) at s==pos.
// One wave per (b, kv, 16-position tile); tiles past pos exit immediately.
// ---------------------------------------------------------------------------
__global__ void scores_wmma(const float* __restrict__ rope, const float* __restrict__ cache_k,
                            const int* __restrict__ pos, float* __restrict__ scores) {
  const int lane = threadIdx.x & 31;
  const int half = lane >> 4;
  const int lm   = lane & 15;
  const int kv   = blockIdx.y;
  const int b    = blockIdx.z;
  const int s0   = blockIdx.x * 16;
  const int cp   = pos[b];
  if (s0 > cp) return;

  const int g = (lm < 4) ? lm : 3;                  // clamp padded A rows
  const float* arow = rope + (size_t)b * (40 * DD) + (size_t)(kv * GG + g) * DD;

  const int s_l = s0 + lm;                          // this lane's position (B: n=lm)
  const int s_c = (s_l > cp) ? cp : s_l;
  const float* krow = (s_c == cp)
      ? (rope + (size_t)b * (40 * DD) + (size_t)(HH + kv) * DD)                  // new k
      : (cache_k + (((size_t)(b * KVH + kv)) * SS + (size_t)s_c) * DD);          // cached k

  v8f c = {};
#pragma unroll 4
  for (int k = 0; k < DD; k += 4) {
    const int ka = k + 2 * half;
    v2f a  = *(const v2f*)(arow + ka);
    v2f bv = *(const v2f*)(krow + ka);              // contiguous float2 along D
    c = wmma_f32_16x16x4(a, bv, c);
  }
  const float scale = 0.08838834764831845f;         // 1/sqrt(128)
  if (half == 0 && s_l <= cp) {
    float* srow = scores + ((size_t)(b * KVH + kv) * GG) * SS;
#pragma unroll
    for (int j = 0; j < 4; ++j) srow[(size_t)j * SS + s_l] = c[j] * scale;
  }
}

// ---------------------------------------------------------------------------
// Per (g, kv, b): max and sum(exp) over s in [0, pos], then rewrite the row
// IN PLACE as normalized probabilities p(s) = exp(score-m)/l.  This keeps the
// transcendental work single-pass (pv_wmma would otherwise redo it 8x).
// ---------------------------------------------------------------------------
__global__ void softmax_normalize(float* __restrict__ scores, const int* __restrict__ pos) {
  const int g = blockIdx.x, kv = blockIdx.y, b = blockIdx.z;
  const int cp = pos[b];
  float* row = scores + (((size_t)(b * KVH + kv)) * GG + g) * SS;
  __shared__ float red[256];
  float m = -3.0e38f;
  for (int s = threadIdx.x; s <= cp; s += 256) m = fmaxf(m, row[s]);
  red[threadIdx.x] = m; __syncthreads();
  for (int off = 128; off > 0; off >>= 1) {
    if ((int)threadIdx.x < off) red[threadIdx.x] = fmaxf(red[threadIdx.x], red[threadIdx.x + off]);
    __syncthreads();
  }
  m = red[0]; __syncthreads();
  float l = 0.0f;
  for (int s = threadIdx.x; s <= cp; s += 256) l += __expf(row[s] - m);
  red[threadIdx.x] = l; __syncthreads();
  for (int off = 128; off > 0; off >>= 1) {
    if ((int)threadIdx.x < off) red[threadIdx.x] += red[threadIdx.x + off];
    __syncthreads();
  }
  const float il = 1.0f / red[0];
  __syncthreads();
  for (int s = threadIdx.x; s <= cp; s += 256) row[s] = __expf(row[s] - m) * il;
}

// ---------------------------------------------------------------------------
// attn[b, kv*G+g, d] = sum_{s<=pos} p(g,s) * V[s,d];  V[pos] = new v.
// WMMA with A = probability quad (K=4 positions), B = V rows.
// 8 waves split S (stride 32 positions); LDS tree-reduce partial tiles.
// ---------------------------------------------------------------------------
__global__ void pv_wmma(const float* __restrict__ probs, const float* __restrict__ cache_v,
                        const float* __restrict__ qkv, const int* __restrict__ pos,
                        float* __restrict__ attn) {
  const int tid  = threadIdx.x;
  const int wave = tid >> 5;                        // 0..7
  const int lane = tid & 31;
  const int half = lane >> 4;
  const int lm   = lane & 15;
  const int d0   = blockIdx.x * 16;
  const int kv   = blockIdx.y;
  const int b    = blockIdx.z;
  const int cp   = pos[b];

  const int g = (lm < 4) ? lm : 3;                  // clamp padded A rows
  const size_t sbase = (((size_t)(b * KVH + kv)) * GG + g) * SS;

  const float* vnew = qkv + (size_t)b * ((HH + 2 * KVH) * DD) + (size_t)(HH + KVH) * DD
                          + (size_t)kv * DD;
  const float* vc   = cache_v + ((size_t)(b * KVH + kv)) * SS * DD;

  v8f c = {};
  for (int s0 = wave * 4; s0 <= cp; s0 += 32) {     // 8 waves x K=4 positions
    const int ka  = 2 * half;
    const int sa0 = s0 + ka, sa1 = sa0 + 1;
    // A: probabilities (zero past pos; clamp load address in-bounds)
    const float p0 = probs[sbase + (size_t)((sa0 > cp) ? cp : sa0)];
    const float p1 = probs[sbase + (size_t)((sa1 > cp) ? cp : sa1)];
    v2f a; a.x = (sa0 <= cp) ? p0 : 0.0f;
           a.y = (sa1 <= cp) ? p1 : 0.0f;
    // B: V rows (new v at s==pos)
    const int sb0 = (sa0 > cp) ? cp : sa0;
    const int sb1 = (sa1 > cp) ? cp : sa1;
    const float* v0 = (sb0 == cp) ? vnew : (vc + (size_t)sb0 * DD);
    const float* v1 = (sb1 == cp) ? vnew : (vc + (size_t)sb1 * DD);
    v2f bv; bv.x = v0[d0 + lm]; bv.y = v1[d0 + lm];
    c = wmma_f32_16x16x4(a, bv, c);
  }

  __shared__ float red[8][32][8];
#pragma unroll
  for (int j = 0; j < 8; ++j) red[wave][lane][j] = c[j];
  __syncthreads();
  if (wave == 0 && half == 0) {
#pragma unroll
    for (int j = 0; j < 4; ++j) {
      float s = 0.0f;
#pragma unroll
      for (int w = 0; w < 8; ++w) s += red[w][lane][j];
      attn[(size_t)b * (HH * DD) + (size_t)(kv * GG + j) * DD + d0 + lm] = s;
    }
  }
}

// ---------------------------------------------------------------------------
extern "C" void kernel_launch(void* const* d_in, const int* in_sizes, int n_in,
                              void* d_out, int out_size, void* d_ws, size_t ws_size,
                              hipStream_t stream) {
  const float* x       = (const float*)d_in[0];   // [4, 4096]
  const float* wqkv    = (const float*)d_in[1];   // [4096, 6144]
  const float* wo      = (const float*)d_in[2];   // [4096, 4096]
  const float* rot     = (const float*)d_in[3];   // [4, 128, 128]
  const float* cache_k = (const float*)d_in[4];   // [4, 8, 8192, 128]  (read-only)
  const float* cache_v = (const float*)d_in[5];   // [4, 8, 8192, 128]  (read-only)
  const int*   cur     = (const int*)d_in[6];     // [4]
  float*       out     = (float*)d_out;           // [4, 4096]

  // Workspace layout (floats). Total ~4.5 MB — L2-resident on MI455X (192 MB L2).
  float* ws        = (float*)d_ws;
  float* qkv_ws    = ws;                              // 4*6144       = 24576
  float* rope_ws   = qkv_ws + (size_t)BB * 6144;      // 4*40*128     = 20480
  float* scores_ws = rope_ws + (size_t)BB * 40 * DD;  // 4*8*4*8192   = 1048576 (+64 pad)
  float* attn_ws   = scores_ws + (size_t)BB * KVH * GG * SS + 64;   // 4*4096 = 16384

  // 1) Fused QKV projection:  qkv_ws[4,6144] = x @ wqkv  (3072 waves)
  gemv_wmma<<<dim3((HH + 2 * KVH) * DD / 16), 256, 0, stream>>>(x, wqkv, qkv_ws, HID,
                                                                (HH + 2 * KVH) * DD);
  // 2) RoPE on q (rows 0..31) and k (rows 32..39)
  rope_wmma<<<dim3(DD / 16, 3, BB), 32, 0, stream>>>(qkv_ws, rot, rope_ws);
  // 3) Attention scores vs cache (+ new k at s==pos), only s <= pos
  scores_wmma<<<dim3(SS / 16, KVH, BB), 32, 0, stream>>>(rope_ws, cache_k, cur, scores_ws);
  // 4) Softmax stats + in-place normalization to probabilities
  softmax_normalize<<<dim3(GG, KVH, BB), 256, 0, stream>>>(scores_ws, cur);
  // 5) P @ V (+ new v at s==pos)
  pv_wmma<<<dim3(DD / 16, KVH, BB), 256, 0, stream>>>(scores_ws, cache_v, qkv_ws,
                                                      cur, attn_ws);
  // 6) Output projection: out = attn @ wo  (2048 waves)
  gemv_wmma<<<dim3(HID / 16), 256, 0, stream>>>(attn_ws, wo, out, HID, HID);
}